// SamplingLayer_45054206935627
// MI455X (gfx1250) — compile-verified
//
#include <hip/hip_runtime.h>

// SamplingLayer: out[b,p,k] = sum_t x[b,p,t] * exp(-((t+1) - w_k)^2 / std^2)
// == GEMM  X[1048576 x 20] * G^T[20 x 16] -> OUT[1048576 x 16]  (all fp32).
// Memory-bound (AI ~= 4.4 flop/byte, ~151 MB total -> ~6.5 us at 23.3 TB/s):
// stream X once (async global->LDS B128 when available), compute with chained
// V_WMMA_F32_16X16X4_F32 (K = 20 = 5 x 4), store D directly.

#define DENSE_L 20
#define L_TILDE 16
// exp(-(d^2)/0.16) == exp(d^2 * -6.25); 6.25 is exactly representable.
#define NEG_INV_STD2 (-6.25f)

typedef __attribute__((ext_vector_type(2))) float v2f;
typedef __attribute__((ext_vector_type(8))) float v8f;

// Async global->LDS builtin takes (addrspace(1) v4i*, addrspace(3) v4i*, imm, imm)
typedef __attribute__((ext_vector_type(4))) int v4i_t;
typedef __attribute__((address_space(1))) v4i_t gv4i;   // pointee type, AS1
typedef __attribute__((address_space(3))) v4i_t lv4i;   // pointee type, AS3

#if __has_builtin(__builtin_amdgcn_global_load_async_to_lds_b128)
#define HAS_ASYNC_LDS 1
#else
#define HAS_ASYNC_LDS 0
#endif

// 256 threads = 8 waves per block; block owns 256 rows (16 row-tiles of 16).
__global__ __launch_bounds__(256)
void sampling_wmma_f32(const float* __restrict__ x,
                       const float* __restrict__ w,
                       float* __restrict__ out) {
  __shared__ float lds[256 * DENSE_L];          // 20 KB

  const int tid  = threadIdx.x;
  const int lane = tid & 31;
  const int wave = tid >> 5;
  const int n     = lane & 15;                  // WMMA column / A-row within tile
  const int hi    = lane >> 4;                  // 0: lanes 0-15, 1: lanes 16-31
  const int t_off = hi << 1;                    // K sub-offset 0 or 2

  const long long blockRow0 = (long long)blockIdx.x * 256;

  // ---- Stage 256 rows x 20 floats into LDS, fully coalesced (B128) ----
  const float4* __restrict__ xv = (const float4*)(x + blockRow0 * DENSE_L);
  float4* lv = (float4*)lds;
#if HAS_ASYNC_LDS
#pragma unroll
  for (int v = 0; v < 5; ++v) {
    const int i = v * 256 + tid;                // 1280 B128 transfers total
    __builtin_amdgcn_global_load_async_to_lds_b128(
        (gv4i*)(xv + i), (lv4i*)(lv + i), /*offset=*/0, /*cpol=*/0);
  }
#else
#pragma unroll
  for (int v = 0; v < 5; ++v) {
    const int i = v * 256 + tid;
    lv[i] = xv[i];
  }
#endif

  // ---- Build B fragments: B[K=20][N=16] = G^T, 5 steps of K=4 ----
  // VGPR0: lanes 0-15 -> K=k0+0, lanes 16-31 -> K=k0+2 ; VGPR1: K=k0+1 / k0+3.
  const float wn = w[n];
  v2f b[5];
#pragma unroll
  for (int j = 0; j < 5; ++j) {
    const float t0 = (float)(4 * j + t_off + 1);   // t runs 1..20
    const float d0 = t0 - wn;
    const float d1 = t0 + 1.0f - wn;
    b[j].x = __expf(d0 * d0 * NEG_INV_STD2);
    b[j].y = __expf(d1 * d1 * NEG_INV_STD2);
  }

#if HAS_ASYNC_LDS
#if __has_builtin(__builtin_amdgcn_s_wait_asynccnt)
  __builtin_amdgcn_s_wait_asynccnt(0);
#else
  asm volatile("s_wait_asynccnt 0x0" ::: "memory");
#endif
#endif
  __syncthreads();

  // ---- Each wave computes 2 output tiles of 16x16 ----
#pragma unroll
  for (int t = 0; t < 2; ++t) {
    const int tile = wave * 2 + t;
    // A fragment source row in LDS for this lane (A 16x4 layout: lane&15 = M,
    // lanes>=16 supply K=2,3 of the same rows). Bank-conflict-free.
    const float* __restrict__ arow = &lds[(tile * 16 + n) * DENSE_L];

    v8f c = {};
#pragma unroll
    for (int j = 0; j < 5; ++j) {
      const v2f a = *(const v2f*)(arow + 4 * j + t_off);   // ds_load_b64
      c = __builtin_amdgcn_wmma_f32_16x16x4_f32(
              /*neg_a=*/false, a, /*neg_b=*/false, b[j],
              /*c_mod=*/(short)0, c, /*reuse_a=*/false, /*reuse_b=*/false);
    }

    // ---- Store D: vgpr r -> row tile*16 + r + 8*hi, col n ----
    const long long row0 = blockRow0 + tile * 16 + (hi << 3);
    float* __restrict__ op = out + row0 * L_TILDE + n;
#pragma unroll
    for (int r = 0; r < 8; ++r) {
      op[r * L_TILDE] = c[r];
    }
  }
}

extern "C" void kernel_launch(void* const* d_in, const int* in_sizes, int n_in,
                              void* d_out, int out_size, void* d_ws, size_t ws_size,
                              hipStream_t stream) {
  const float* x   = (const float*)d_in[0];   // [4096, 5120] fp32
  const float* w   = (const float*)d_in[1];   // [16] fp32
  float*       out = (float*)d_out;           // [4096, 4096] fp32

  const int rows   = in_sizes[0] / DENSE_L;   // 1,048,576
  const int blocks = rows / 256;              // 4096 blocks x 256 threads
  sampling_wmma_f32<<<blocks, 256, 0, stream>>>(x, w, out);
}